// AttentionLayerOrtho_16174846836815
// MI455X (gfx1250) — compile-verified
//
#include <hip/hip_runtime.h>
#include <hip/hip_bf16.h>
#include <math.h>
#include <stdint.h>

typedef float v2f __attribute__((ext_vector_type(2)));
typedef float v8f __attribute__((ext_vector_type(8)));

#define NB  16    // batch
#define TE  128   // encoder timesteps
#define TD  128   // decoder timesteps
#define H   256   // hidden
#define DT  8     // decoder rows per block in stage 4

// ---------------------------------------------------------------------------
// CDNA5 transcendental helpers: V_TANH_F32 / V_EXP_F32 on the TRANS unit.
// ---------------------------------------------------------------------------
__device__ __forceinline__ float fast_tanh(float x) {
#if __has_builtin(__builtin_amdgcn_tanhf)
    return __builtin_amdgcn_tanhf(x);
#elif __has_builtin(__builtin_amdgcn_tanh_f32)
    return __builtin_amdgcn_tanh_f32(x);
#else
    return tanhf(x);
#endif
}

__device__ __forceinline__ float fast_exp(float x) {
#if __has_builtin(__builtin_amdgcn_exp2f)
    return __builtin_amdgcn_exp2f(x * 1.4426950408889634f);  // e^x = 2^(x*log2e)
#else
    return expf(x);
#endif
}

// ---------------------------------------------------------------------------
// CDNA5 async global->LDS copy (16B per lane), tracked by ASYNCcnt.
// Inline asm per bridge doc (portable across toolchains). The ISA guarantees
// generic shared addresses map to LDS offsets via addr[31:0].
// ---------------------------------------------------------------------------
__device__ __forceinline__ void async_copy_b128(void* lds_dst, const void* gsrc) {
    uint32_t l = (uint32_t)(uintptr_t)lds_dst;
    uint64_t g = (uint64_t)(uintptr_t)gsrc;
    asm volatile("global_load_async_to_lds_b128 %0, %1, off"
                 :: "v"(l), "v"(g) : "memory");
}

__device__ __forceinline__ void wait_async0() {
#if __has_builtin(__builtin_amdgcn_s_wait_asynccnt)
    __builtin_amdgcn_s_wait_asynccnt(0);
#else
    asm volatile("s_wait_asynccnt 0" ::: "memory");
#endif
}

// ---------------------------------------------------------------------------
// Stage 1: Gram-Schmidt-style orthogonalization (faithful to reference:
// out = x - ((x*s)/(x*x))*x with s = exclusive cumsum over encoder time).
// ---------------------------------------------------------------------------
__global__ void ortho_kernel(const float* __restrict__ enc,
                             float* __restrict__ out) {
    int b = blockIdx.x;         // 0..NB-1
    int h = threadIdx.x;        // 0..H-1
    const float* ep = enc + (size_t)b * TE * H + h;
    float*       op = out + (size_t)b * TE * H + h;
    float s = 0.0f;
    for (int t = 0; t < TE; ++t) {
        float x = ep[(size_t)t * H];
        float o = x - ((x * s) / (x * x)) * x;   // t=0: s=0 -> o=x
        op[(size_t)t * H] = o;
        s += x;
    }
}

// ---------------------------------------------------------------------------
// fp32 GEMM using V_WMMA_F32_16X16X4_F32 (wave32).
// One wave per 16x32 C region (two 16x16 tiles sharing the A fragment).
// grid=(N/32, M/16, batch).
// ---------------------------------------------------------------------------
__global__ void gemm16_wmma_f32(const float* __restrict__ A,
                                const float* __restrict__ Bm,
                                float* __restrict__ C,
                                int K, int lda, int ldb, int ldc,
                                long long sA, long long sB, long long sC) {
    const float* Ab = A  + (size_t)blockIdx.z * sA;
    const float* Bb = Bm + (size_t)blockIdx.z * sB;
    float*       Cb = C  + (size_t)blockIdx.z * sC;

    int lane = threadIdx.x;      // block = 32 threads = 1 wave
    int l16  = lane & 15;
    int kh   = lane >> 4;        // which K-half (0/1) this lane supplies
    int m0   = blockIdx.y * 16;
    int n0   = blockIdx.x * 32;  // two adjacent 16-wide tiles

    const float* ap = Ab + (size_t)(m0 + l16) * lda + 2 * kh;
    const float* bp = Bb + (size_t)(2 * kh) * ldb + (n0 + l16);

    __builtin_prefetch(ap, 0, 0);
    __builtin_prefetch(bp, 0, 0);
    __builtin_prefetch(bp + 16, 0, 0);

    v8f acc0 = {};
    v8f acc1 = {};
#pragma unroll 4
    for (int k = 0; k < K; k += 4) {
        v2f a, b0, b1;
        a.x  = ap[0];
        a.y  = ap[1];
        b0.x = bp[0];
        b0.y = bp[ldb];
        b1.x = bp[16];
        b1.y = bp[ldb + 16];
        acc0 = __builtin_amdgcn_wmma_f32_16x16x4_f32(
            false, a, false, b0, (short)0, acc0, false, false);
        acc1 = __builtin_amdgcn_wmma_f32_16x16x4_f32(
            false, a, false, b1, (short)0, acc1, false, false);
        ap += 4;
        bp += (size_t)4 * ldb;
    }

    int row = m0 + (kh << 3);
    int col = n0 + l16;
#pragma unroll
    for (int r = 0; r < 8; ++r) {
        Cb[(size_t)(row + r) * ldc + col]      = acc0[r];
        Cb[(size_t)(row + r) * ldc + col + 16] = acc1[r];
    }
}

// ---------------------------------------------------------------------------
// Stage 4: energy[b,d,e] = sum_h tanh(was[b,e,h] + uah[b,d,h]) * V[h], then
// softmax over e. One 256-thread block per (b, 8 decoder rows): each 8-row
// `was` tile (8KB contiguous) is async-copied into LDS ONCE and shared by all
// 8 waves (8x L2-traffic cut vs block-per-d). Each wave owns one d: its uah
// row and V_a live in registers; its 128 energies live in 4 registers via
// all-lanes shfl_xor reductions; softmax is fully wave-local.
// ---------------------------------------------------------------------------
__global__ void __launch_bounds__(256)
energy_softmax_kernel(const float* __restrict__ was,  // [B,TE,H]
                      const float* __restrict__ uah,  // [B,TD,H]
                      const float* __restrict__ Va,   // [H]
                      float* __restrict__ eout) {     // [B,TD,TE]
    __shared__ float ws_s[DT * H];   // 8 rows of was, 8KB

    int bx   = blockIdx.x;           // NB * (TD/DT) blocks
    int b    = bx / (TD / DT);
    int d0   = (bx % (TD / DT)) * DT;
    int tid  = threadIdx.x;          // 0..255
    int wave = tid >> 5;             // 0..7  -> d = d0 + wave
    int lane = tid & 31;
    int d    = d0 + wave;
    int h0   = lane * 8;             // 8 contiguous h per lane

    // Per-wave constants in registers: uah row for this d, and V_a.
    const float* up = uah + ((size_t)b * TD + d) * H + h0;
    float4 u0 = *(const float4*)(up);
    float4 u1 = *(const float4*)(up + 4);
    float4 v0 = *(const float4*)(Va + h0);
    float4 v1 = *(const float4*)(Va + h0 + 4);

    const float* wb = was + (size_t)b * TE * H;
    float en[4];                     // energy for e = 32*q + lane

#pragma unroll 1
    for (int et = 0; et < TE / DT; ++et) {
        // Async-stage the 8-row (contiguous 8KB) was tile into LDS.
        const float* gsrc = wb + (size_t)et * DT * H;
        async_copy_b128(&ws_s[tid * 4],        gsrc + tid * 4);
        async_copy_b128(&ws_s[1024 + tid * 4], gsrc + 1024 + tid * 4);
        wait_async0();
        __syncthreads();

#pragma unroll
        for (int i = 0; i < DT; ++i) {
            const float* wrow = &ws_s[i * H + h0];
            float4 w0 = *(const float4*)(wrow);
            float4 w1 = *(const float4*)(wrow + 4);
            float p = 0.0f;
            p += fast_tanh(w0.x + u0.x) * v0.x;
            p += fast_tanh(w0.y + u0.y) * v0.y;
            p += fast_tanh(w0.z + u0.z) * v0.z;
            p += fast_tanh(w0.w + u0.w) * v0.w;
            p += fast_tanh(w1.x + u1.x) * v1.x;
            p += fast_tanh(w1.y + u1.y) * v1.y;
            p += fast_tanh(w1.z + u1.z) * v1.z;
            p += fast_tanh(w1.w + u1.w) * v1.w;
#pragma unroll
            for (int off = 16; off > 0; off >>= 1)
                p += __shfl_xor(p, off, 32);     // all lanes hold full sum
            int e = et * DT + i;
            if (lane == (e & 31)) en[e >> 5] = p;
        }
        __syncthreads();             // protect LDS tile before next overwrite
    }

    // Wave-local softmax over the 128 energies (lane holds e = 32q+lane).
    float m = fmaxf(fmaxf(en[0], en[1]), fmaxf(en[2], en[3]));
#pragma unroll
    for (int off = 16; off > 0; off >>= 1)
        m = fmaxf(m, __shfl_xor(m, off, 32));
    float ex[4];
    float s = 0.0f;
#pragma unroll
    for (int q = 0; q < 4; ++q) {
        ex[q] = fast_exp(en[q] - m);
        s += ex[q];
    }
#pragma unroll
    for (int off = 16; off > 0; off >>= 1)
        s += __shfl_xor(s, off, 32);

    float* op = eout + ((size_t)b * TD + d) * TE;
#pragma unroll
    for (int q = 0; q < 4; ++q)
        op[32 * q + lane] = ex[q] / s;
}

// ---------------------------------------------------------------------------
extern "C" void kernel_launch(void* const* d_in, const int* in_sizes, int n_in,
                              void* d_out, int out_size, void* d_ws, size_t ws_size,
                              hipStream_t stream) {
    const float* enc = (const float*)d_in[0];  // [16,128,256]
    const float* dec = (const float*)d_in[1];  // [16,128,256]
    const float* Wa  = (const float*)d_in[2];  // [256,256]
    const float* Ua  = (const float*)d_in[3];  // [256,256]
    const float* Va  = (const float*)d_in[4];  // [256,1]

    float* c_out = (float*)d_out;                       // [16,128,256]
    float* e_out = c_out + (size_t)NB * TD * H;         // [16,128,128]

    float* ortho = (float*)d_ws;                        // [16,128,256]
    float* was   = ortho + (size_t)NB * TE * H;         // [16,128,256]
    float* uah   = was   + (size_t)NB * TE * H;         // [16,128,256]

    // Stage 1: orthogonalize encoder outputs
    ortho_kernel<<<NB, H, 0, stream>>>(enc, ortho);

    // Stage 2: was = ortho @ W_a  (flattened M = B*TE = 2048, N = K = 256)
    gemm16_wmma_f32<<<dim3(H / 32, (NB * TE) / 16, 1), 32, 0, stream>>>(
        ortho, Wa, was, /*K=*/H, /*lda=*/H, /*ldb=*/H, /*ldc=*/H, 0, 0, 0);

    // Stage 3: uah = dec @ U_a
    gemm16_wmma_f32<<<dim3(H / 32, (NB * TD) / 16, 1), 32, 0, stream>>>(
        dec, Ua, uah, /*K=*/H, /*lda=*/H, /*ldb=*/H, /*ldc=*/H, 0, 0, 0);

    // Stage 4: energies + softmax -> e_out
    energy_softmax_kernel<<<NB * (TD / DT), 256, 0, stream>>>(was, uah, Va, e_out);

    // Stage 5: c_out[b] = e_out[b] @ ortho[b]  (batched 128x256x128 GEMM)
    gemm16_wmma_f32<<<dim3(H / 32, TD / 16, NB), 32, 0, stream>>>(
        e_out, ortho, c_out, /*K=*/TE, /*lda=*/TE, /*ldb=*/H, /*ldc=*/H,
        (long long)TD * TE, (long long)TE * H, (long long)TD * H);
}